// GraphAttention_76020921139772
// MI455X (gfx1250) — compile-verified
//
#include <hip/hip_runtime.h>
#include <math.h>

#define BB 8
#define NN 2048
#define FF 128

typedef __attribute__((ext_vector_type(2))) float v2f;
typedef __attribute__((ext_vector_type(8))) float v8f;

// ---------------- K0: zero-init the atomic accumulators ----------------
__global__ void k0_zero(float* __restrict__ s_i, float* __restrict__ colsum) {
  int i = blockIdx.x * blockDim.x + threadIdx.x;
  if (i < BB * NN) s_i[i] = 0.f;
  if (i < BB * FF) colsum[i] = 0.f;
}

// ---------------- K1: fp32 WMMA GEMM (XW = X@W), fused reductions ------
// Each wave computes one 16x16 tile of XW[b]. Block = 8 waves = 16 rows x 128 cols.
// Tile is immediately reduced into:
//   s_i[b,row]   += sum_col XW[row,col] * a_i[col]   (atomic)
//   colsum[b,col]+= sum_row XW[row,col]              (atomic)
// XW itself is never stored.
__global__ __launch_bounds__(256) void k1_gemm(const float* __restrict__ X,
                                               const float* __restrict__ W,
                                               const float* __restrict__ a,
                                               float* __restrict__ s_i,
                                               float* __restrict__ colsum) {
  const int lane = threadIdx.x & 31;
  const int wave = threadIdx.x >> 5;     // colTile 0..7
  const int m    = lane & 15;
  const int half = lane >> 4;            // 0: K0/K1 rows M0-7 of D; 1: K2/K3 rows M8-15
  const int b       = blockIdx.x >> 7;   // / (N/16)
  const int rowTile = blockIdx.x & 127;
  const int r0  = rowTile * 16;
  const int c0  = wave * 16;
  const int col = c0 + m;
  const int row = r0 + m;

  const float* Xrow = X + ((size_t)b * NN + row) * FF;

  v8f acc = {0.f, 0.f, 0.f, 0.f, 0.f, 0.f, 0.f, 0.f};

  #pragma unroll 4
  for (int k = 0; k < FF; k += 4) {
    const int kk = k + half * 2;
    // A fragment: 16x4 fp32. Lane holds X[row, kk], X[row, kk+1] (contiguous, 8B aligned).
    v2f afrag = *(const v2f*)(Xrow + kk);
    // B fragment: 4x16 fp32. Lane holds W[kk, col], W[kk+1, col].
    v2f bfrag;
    bfrag.x = W[kk * FF + col];
    bfrag.y = W[(kk + 1) * FF + col];
    acc = __builtin_amdgcn_wmma_f32_16x16x4_f32(false, afrag, false, bfrag,
                                                (short)0, acc, false, false);
  }

  // ---- colsum: per-lane column partial over its 8 rows, combine halves ----
  float cs = acc[0] + acc[1] + acc[2] + acc[3] + acc[4] + acc[5] + acc[6] + acc[7];
  cs += __shfl_xor(cs, 16);              // rows 0-7 + rows 8-15 (same col)
  if (half == 0) atomicAdd(&colsum[b * FF + col], cs);

  // ---- s_i partial: dot tile columns with a_i over this 16-col slice ----
  const float ai = a[FF + col];          // a_i = a[0, F:]
  #pragma unroll
  for (int v = 0; v < 8; ++v) {
    float p = acc[v] * ai;
    p += __shfl_xor(p, 1);
    p += __shfl_xor(p, 2);
    p += __shfl_xor(p, 4);
    p += __shfl_xor(p, 8);               // reduce across the 16 lanes of this half
    if (m == 0) atomicAdd(&s_i[b * NN + r0 + half * 8 + v], p);
  }
}

// ---------------- K2: e = exp(s_i) -------------------------------------
__global__ void k2_exp(const float* __restrict__ s_i, float* __restrict__ e) {
  int i = blockIdx.x * blockDim.x + threadIdx.x;
  if (i < BB * NN) e[i] = expf(s_i[i]);
}

// ---------------- K3: r = A @ e ; scale = (r!=0) ? e/r : 0 --------------
// One wave per output row; float4 streaming over a 2048-float row of A.
__global__ __launch_bounds__(256) void k3_rscale(const float* __restrict__ A,
                                                 const float* __restrict__ e,
                                                 float* __restrict__ scale) {
  const int lane = threadIdx.x & 31;
  const int wave = threadIdx.x >> 5;
  const int g = blockIdx.x * 8 + wave;   // row id in [0, B*N)
  const int b = g >> 11;
  const int i = g & (NN - 1);
  const float4* Ar = (const float4*)(A + ((size_t)b * NN + i) * NN);
  const float4* er = (const float4*)(e + b * NN);
  float sum = 0.f;
  #pragma unroll 4
  for (int j = lane; j < NN / 4; j += 32) {
    float4 av = Ar[j];
    float4 ev = er[j];
    sum += av.x * ev.x + av.y * ev.y + av.z * ev.z + av.w * ev.w;
  }
  sum += __shfl_xor(sum, 1);
  sum += __shfl_xor(sum, 2);
  sum += __shfl_xor(sum, 4);
  sum += __shfl_xor(sum, 8);
  sum += __shfl_xor(sum, 16);
  if (lane == 0) {
    float ei = e[b * NN + i];
    scale[g] = (sum != 0.f) ? (ei / sum) : 0.f;
  }
}

// ---------------- K4: t = A @ scale ; H = t*colsum + bias_W -------------
__global__ __launch_bounds__(256) void k4_out(const float* __restrict__ A,
                                              const float* __restrict__ scale,
                                              const float* __restrict__ colsum,
                                              const float* __restrict__ biasW,
                                              float* __restrict__ H) {
  const int lane = threadIdx.x & 31;
  const int wave = threadIdx.x >> 5;
  const int g = blockIdx.x * 8 + wave;
  const int b = g >> 11;
  const int i = g & (NN - 1);
  const float4* Ar = (const float4*)(A + ((size_t)b * NN + i) * NN);
  const float4* sr = (const float4*)(scale + b * NN);
  float sum = 0.f;
  #pragma unroll 4
  for (int j = lane; j < NN / 4; j += 32) {
    float4 av = Ar[j];
    float4 sv = sr[j];
    sum += av.x * sv.x + av.y * sv.y + av.z * sv.z + av.w * sv.w;
  }
  sum += __shfl_xor(sum, 1);
  sum += __shfl_xor(sum, 2);
  sum += __shfl_xor(sum, 4);
  sum += __shfl_xor(sum, 8);
  sum += __shfl_xor(sum, 16);            // all lanes now hold t
  const float t = sum;
  const float4 c4 = ((const float4*)(colsum + b * FF))[lane]; // 32 lanes * float4 = 128
  const float4 b4 = ((const float4*)biasW)[lane];
  float4 o;
  o.x = t * c4.x + b4.x;
  o.y = t * c4.y + b4.y;
  o.z = t * c4.z + b4.z;
  o.w = t * c4.w + b4.w;
  ((float4*)(H + ((size_t)b * NN + i) * FF))[lane] = o;
}

extern "C" void kernel_launch(void* const* d_in, const int* in_sizes, int n_in,
                              void* d_out, int out_size, void* d_ws, size_t ws_size,
                              hipStream_t stream) {
  const float* X     = (const float*)d_in[0];  // (B,N,F_IN)
  const float* A     = (const float*)d_in[1];  // (B,N,N)
  const float* W     = (const float*)d_in[2];  // (F_IN,F_OUT)
  const float* a     = (const float*)d_in[3];  // (1, 2*F_OUT)
  // d_in[4] = bias_a : algebraically cancels out of the output
  const float* biasW = (const float*)d_in[5];  // (F_OUT,)
  float* H = (float*)d_out;

  float* s_i    = (float*)d_ws;                // B*N
  float* e      = s_i + BB * NN;               // B*N
  float* scale  = e + BB * NN;                 // B*N
  float* colsum = scale + BB * NN;             // B*F

  // XW is never materialized; total ws usage ~200 KB.
  k0_zero<<<(BB * NN + 255) / 256, 256, 0, stream>>>(s_i, colsum);
  k1_gemm<<<BB * (NN / 16), 256, 0, stream>>>(X, W, a, s_i, colsum);
  k2_exp<<<(BB * NN + 255) / 256, 256, 0, stream>>>(s_i, e);
  k3_rscale<<<(BB * NN) / 8, 256, 0, stream>>>(A, e, scale);
  k4_out<<<(BB * NN) / 8, 256, 0, stream>>>(A, scale, colsum, biasW, H);
}